// FixedFanInCuda_13597866459292
// MI455X (gfx1250) — compile-verified
//
#include <hip/hip_runtime.h>

// Fixed-fan-in sparse linear for MI455X (gfx1250, wave32).
//   out[b,o] = sum_k input[b, mask[o,k]] * W[o,k] + bias[o]
//
// Roofline: 2.15 GFLOP but 4.3G random 4B gathers of input -> LDS-gather
// bound, not HBM bound. WMMA cannot express the per-output random gather
// without 16x wasted MACs, so the matrix path is intentionally unused; the
// CDNA5 features that pay off are the 320KB LDS (16-row x 4096-col fp32 tile
// resident per WGP), async global->LDS loads (per-lane LDS destination =>
// free transpose during fill, ASYNCcnt-tracked), and b128 LDS gathers
// (128 gathered elements per wave ds op).

#define NB      4096   // batch rows
#define IND     4096   // input features
#define OUTD    4096   // output features
#define FAN     64     // fixed fan-in

#define BT        16   // batch rows resident in LDS per workgroup
#define COLSTRIDE 16   // dwords per LDS column (16B-aligned starts, 256 KB
                       // total: safe margin under the 320KB/WG cap, and
                       // gather address folds to a single lshl_add)
#define THREADS   1024
#define NWAVES    (THREADS / 32)
#define OTILE     8    // output columns per wave tile (8 groups x 4 lanes)

__global__ __launch_bounds__(THREADS, 1)
void ffi_sparse_linear(const float* __restrict__ input,   // [NB][IND]
                       const float* __restrict__ weight,  // [OUTD][FAN]
                       const int*   __restrict__ mask,    // [OUTD][FAN]
                       const float* __restrict__ bias,    // [OUTD]
                       float*       __restrict__ out)     // [NB][OUTD]
{
    extern __shared__ float lds[];          // IND * COLSTRIDE dwords = 256 KB

    const int tid = threadIdx.x;
    const int b0  = blockIdx.x * BT;        // this WG's batch tile

    // ---------------------------------------------------------------
    // Phase 1: async transposing fill.
    //   lds[c*COLSTRIDE + b] = input[b0+b][c]
    // Thread t -> (b = t/64, c = (t%64) + 64*i). Each wave has constant b and
    // 32 consecutive c => 128B-coalesced global reads. The async-to-LDS op
    // takes a per-lane LDS destination address, so the transpose is free.
    // Aggregate LDS write banks = 16*(c%4) + b cover all 64 banks across the
    // workgroup's 16 b values.
    // ---------------------------------------------------------------
    {
        const int b  = tid >> 6;            // 0..15
        const int c0 = tid & 63;            // 0..63
        const float* gsrc = input + (size_t)(b0 + b) * IND + c0;
        // addrspace(3) offset: low 32 bits of the generic LDS pointer
        unsigned loff = (unsigned)(reinterpret_cast<uintptr_t>(
                            lds + ((unsigned)c0 << 4) + (unsigned)b));
        #pragma unroll 8
        for (int i = 0; i < IND / 64; ++i) {
            asm volatile("global_load_async_to_lds_b32 %0, %1, off"
                         :: "v"(loff), "v"(gsrc)
                         : "memory");
            gsrc += 64;
            loff += 64u * COLSTRIDE * 4u;   // step c by 64 columns (4 KB)
        }
        // wait for this wave's async transfers to land in LDS
        asm volatile("s_wait_asynccnt 0x0" ::: "memory");
    }
    __syncthreads();   // all waves' fills visible

    // ---------------------------------------------------------------
    // Phase 2: gather + FMA.
    // Wave tile = 8 outputs x 16 batch rows. Lane = (g = lane/4 -> output
    // o = tile*8+g, q = lane%4 -> batch quad b_local = 4q..4q+3).
    // Per k: one ds_load_b128 per lane pulls 4 batch values of gathered
    // column mask[o][k]; the whole wave moves 128 gathered elements per ds op.
    // ---------------------------------------------------------------
    const int wave = tid >> 5;
    const int lane = tid & 31;
    const int g    = lane >> 2;                         // 0..7
    const unsigned byte_q = (unsigned)(lane & 3) * 16u; // 16B-aligned quad
    const char* lbase = reinterpret_cast<const char*>(lds);

    for (int tileo = wave; tileo < OUTD / OTILE; tileo += NWAVES) {
        const int o = tileo * OTILE + g;              // this lane's output col
        const int4*   mrow = reinterpret_cast<const int4*>(mask + o * FAN);
        const float4* wrow = reinterpret_cast<const float4*>(weight + o * FAN);

        float a0 = 0.f, a1 = 0.f, a2 = 0.f, a3 = 0.f;

        #pragma unroll 4
        for (int kc = 0; kc < FAN / 4; ++kc) {
            const int4   m4 = mrow[kc];               // 4 gather indices (L2-hot)
            const float4 w4 = wrow[kc];               // 4 weights
            {
                const float4 v = *reinterpret_cast<const float4*>(
                    lbase + ((unsigned)m4.x << 6) + byte_q);
                a0 = fmaf(v.x, w4.x, a0); a1 = fmaf(v.y, w4.x, a1);
                a2 = fmaf(v.z, w4.x, a2); a3 = fmaf(v.w, w4.x, a3);
            }
            {
                const float4 v = *reinterpret_cast<const float4*>(
                    lbase + ((unsigned)m4.y << 6) + byte_q);
                a0 = fmaf(v.x, w4.y, a0); a1 = fmaf(v.y, w4.y, a1);
                a2 = fmaf(v.z, w4.y, a2); a3 = fmaf(v.w, w4.y, a3);
            }
            {
                const float4 v = *reinterpret_cast<const float4*>(
                    lbase + ((unsigned)m4.z << 6) + byte_q);
                a0 = fmaf(v.x, w4.z, a0); a1 = fmaf(v.y, w4.z, a1);
                a2 = fmaf(v.z, w4.z, a2); a3 = fmaf(v.w, w4.z, a3);
            }
            {
                const float4 v = *reinterpret_cast<const float4*>(
                    lbase + ((unsigned)m4.w << 6) + byte_q);
                a0 = fmaf(v.x, w4.w, a0); a1 = fmaf(v.y, w4.w, a1);
                a2 = fmaf(v.z, w4.w, a2); a3 = fmaf(v.w, w4.w, a3);
            }
        }

        const float bo = bias[o];
        float* op = out + (size_t)(b0 + (lane & 3) * 4) * OUTD + o;
        op[0 * (size_t)OUTD] = a0 + bo;
        op[1 * (size_t)OUTD] = a1 + bo;
        op[2 * (size_t)OUTD] = a2 + bo;
        op[3 * (size_t)OUTD] = a3 + bo;
    }
}

extern "C" void kernel_launch(void* const* d_in, const int* in_sizes, int n_in,
                              void* d_out, int out_size, void* d_ws, size_t ws_size,
                              hipStream_t stream) {
    (void)in_sizes; (void)n_in; (void)out_size; (void)d_ws; (void)ws_size;

    const float* input = (const float*)d_in[0];   // [4096,4096] f32
    const float* w     = (const float*)d_in[1];   // [4096,64]   f32
    const int*   mask  = (const int*)  d_in[2];   // [4096,64]   i32
    const float* bias  = (const float*)d_in[3];   // [4096]      f32
    float*       out   = (float*)d_out;           // [4096,4096] f32

    const dim3 grid(NB / BT);                     // 256 workgroups (1 per WGP)
    const dim3 block(THREADS);                    // 32 waves (wave32)
    const size_t lds_bytes = (size_t)IND * COLSTRIDE * sizeof(float); // 256 KB

    hipLaunchKernelGGL(ffi_sparse_linear, grid, block, lds_bytes, stream,
                       input, w, mask, bias, out);
}